// HierarchicalLLLoss_6476810682431
// MI455X (gfx1250) — compile-verified
//
#include <hip/hip_runtime.h>
#include <math.h>

// HierarchicalLLLoss on MI455X (gfx1250).
// Only `inputs` (16 MiB) must move: HBM-bound, ~0.72 us floor at 23.3 TB/s.
// onehot_num / onehot_den are never read (tree structure is implicit).
// Per wave: async-DMA one 4KB row into LDS, reduce it through the WMMA pipe
// (B = ones => row sums of A), then one lane does the 10 log-ratios.

#define NUM_CLASSES 1024
#define DEPTH 10
#define WAVES_PER_BLOCK 8
// per-wave LDS floats: row 1024 + level2 nodes 256 + level4 64 + level6 16
#define WAVE_LDS (1024 + 256 + 64 + 16)

typedef float v2f __attribute__((ext_vector_type(2)));
typedef float v4f __attribute__((ext_vector_type(4)));
typedef float v8f __attribute__((ext_vector_type(8)));

// 64 contiguous floats in src -> 16 sums-of-4 in dst, via V_WMMA_F32_16X16X4_F32.
// A(16x4) layout: lanes 0-15 hold A[m][0..1], lanes 16-31 hold A[m][2..3] (m=lane&15).
// B = all ones (layout-independent), C = 0  =>  D[m][n] = sum_k A[m][k] for every n.
// D layout: VGPR v: lanes 0-15 -> M=v, lanes 16-31 -> M=v+8; column 0 lives in
// lanes 0 and 16, so those two lanes write the 16 node values with 2x ds_store_b128.
__device__ __forceinline__ void wmma_reduce64(const float* src, float* dst, int lane) {
  const int idx = ((lane & 15) << 2) | ((lane >> 4) << 1);  // 4*m + koff, even
  v2f a = *(const v2f*)(src + idx);                          // ds_load_b64
  v2f bones = {1.0f, 1.0f};
  v8f c = {0.f, 0.f, 0.f, 0.f, 0.f, 0.f, 0.f, 0.f};
  v8f d = __builtin_amdgcn_wmma_f32_16x16x4_f32(false, a, false, bones,
                                                (short)0, c, false, false);
  if ((lane & 15) == 0) {                 // lanes 0 and 16 only, one predicate
    float* p = dst + ((lane >> 4) << 3);  // lane0 -> rows 0..7, lane16 -> rows 8..15
    v4f lo = {d[0], d[1], d[2], d[3]};
    v4f hi = {d[4], d[5], d[6], d[7]};
    *(v4f*)p = lo;                        // ds_store_b128
    *(v4f*)(p + 4) = hi;                  // ds_store_b128
  }
  __builtin_amdgcn_wave_barrier();
}

__global__ void __launch_bounds__(256)
hll_per_sample(const float* __restrict__ inputs, const int* __restrict__ target,
               const float* __restrict__ weights, float* __restrict__ per_sample,
               int batch) {
  __shared__ __align__(16) float lds[WAVES_PER_BLOCK * WAVE_LDS];
  const int lane = threadIdx.x & 31;
  const int w = threadIdx.x >> 5;
  const int b = blockIdx.x * WAVES_PER_BLOCK + w;   // wave-uniform
  if (b >= batch) return;                            // whole wave exits together

  float* R  = &lds[w * WAVE_LDS];   // row: 1024 f32
  float* N2 = R + 1024;             // level-2 tree nodes: 256
  float* N4 = N2 + 256;             // level-4 nodes: 64
  float* N6 = N4 + 64;              // level-6 nodes: 16

  // ---- async DMA: row b (4 KB) global -> LDS, 8 x b128 per wave ----
  unsigned lds_base = (unsigned)(unsigned long long)R;  // low 32 bits = LDS offset
  unsigned long long gbase = (unsigned long long)b * (NUM_CLASSES * 4ull);
#pragma unroll
  for (int i = 0; i < 8; ++i) {
    unsigned lds_addr = lds_base + (unsigned)((i * 32 + lane) * 16);
    unsigned goff = (unsigned)(gbase + (unsigned long long)(i * 32 + lane) * 16ull);
    asm volatile("global_load_async_to_lds_b128 %0, %1, %2"
                 :: "v"(lds_addr), "v"(goff), "s"(inputs) : "memory");
  }
  asm volatile("s_wait_asynccnt 0x0" ::: "memory");
  __builtin_amdgcn_wave_barrier();

  // ---- tree reduction on the matrix pipe: 1024 -> 256 -> 64 -> 16 nodes ----
#pragma unroll
  for (int c = 0; c < 16; ++c) wmma_reduce64(R + c * 64, N2 + c * 16, lane);
#pragma unroll
  for (int c = 0; c < 4; ++c)  wmma_reduce64(N2 + c * 64, N4 + c * 16, lane);
  wmma_reduce64(N4, N6, lane);

  // ---- path sums s_0..s_10 around target leaf, then weighted log-ratios ----
  if (lane == 0) {
    const int t = target[b];
    float s[DEPTH + 1];
    s[0] = R[t];
    s[1] = R[t & ~1] + R[(t & ~1) | 1];
    s[2] = N2[t >> 2];
    { int q = (t >> 3) << 1; s[3] = N2[q] + N2[q + 1]; }
    s[4] = N4[t >> 4];
    { int q = (t >> 5) << 1; s[5] = N4[q] + N4[q + 1]; }
    s[6] = N6[t >> 6];
    { int q = (t >> 7) << 1; s[7] = N6[q] + N6[q + 1]; }
    { int q = (t >> 8) << 2; s[8] = (N6[q] + N6[q + 1]) + (N6[q + 2] + N6[q + 3]); }
    { int q = (t >> 9) << 3;
      float a0 = 0.f;
#pragma unroll
      for (int i = 0; i < 8; ++i) a0 += N6[q + i];
      s[9] = a0; }
    { float a0 = 0.f;
#pragma unroll
      for (int i = 0; i < 16; ++i) a0 += N6[i];
      s[10] = a0; }

    float acc = 0.f;
#pragma unroll
    for (int j = 0; j < DEPTH; ++j) {
      float num = s[j], den = s[j + 1];
      float val = (num != 0.0f) ? -logf(num / den) : 0.0f;  // matches where() guard
      acc += weights[t * DEPTH + j] * val;
    }
    per_sample[b] = acc;
  }
}

// deterministic batch mean: single block, fixed tree order
__global__ void __launch_bounds__(256)
hll_mean(const float* __restrict__ ws, float* __restrict__ out, int n) {
  __shared__ float sm[256];
  float s = 0.f;
  for (int i = threadIdx.x; i < n; i += 256) s += ws[i];
  sm[threadIdx.x] = s;
  __syncthreads();
  for (int off = 128; off > 0; off >>= 1) {
    if ((int)threadIdx.x < off) sm[threadIdx.x] += sm[threadIdx.x + off];
    __syncthreads();
  }
  if (threadIdx.x == 0) out[0] = sm[0] / (float)n;
}

extern "C" void kernel_launch(void* const* d_in, const int* in_sizes, int n_in,
                              void* d_out, int out_size, void* d_ws, size_t ws_size,
                              hipStream_t stream) {
  // setup_inputs order: inputs, target, onehot_num (unused), onehot_den (unused), weights
  const float* inputs  = (const float*)d_in[0];
  const int*   target  = (const int*)d_in[1];
  const float* weights = (const float*)d_in[4];
  float* ws  = (float*)d_ws;   // per-sample losses: batch f32
  float* out = (float*)d_out;  // scalar mean

  const int batch = in_sizes[1];
  const int blocks = (batch + WAVES_PER_BLOCK - 1) / WAVES_PER_BLOCK;
  hll_per_sample<<<blocks, 256, 0, stream>>>(inputs, target, weights, ws, batch);
  hll_mean<<<1, 256, 0, stream>>>(ws, out, batch);
}